// FixedConvQuadInterp3d_32710470926437
// MI455X (gfx1250) — compile-verified
//
#include <hip/hip_runtime.h>
#include <stdint.h>

namespace {

constexpr int B_ = 2, C_ = 8, D_ = 4, H_ = 256, W_ = 384;
constexpr int TILE_W = 128;
constexpr float BONUS = 10.0f;   // STRICT_MAXIMA_BONUS
constexpr float ADJ   = 0.0f;    // FINAL_ADJUSTMENT

__device__ __forceinline__ void async_load_f32_to_lds(const float* gaddr,
                                                      const float* lds_generic) {
  // LDS aperture: low 32 bits of the generic shared pointer are the
  // workgroup-relative LDS byte offset (ISA 10.2).
  unsigned lds_off = (unsigned)(uintptr_t)lds_generic;
  // GLOBAL_LOAD_ASYNC_TO_LDS_B32 (tracked by ASYNCcnt, no VGPR staging)
  asm volatile("global_load_async_to_lds_b32 %0, %1, off"
               :
               : "v"(lds_off), "v"(gaddr)
               : "memory");
}

__global__ __launch_bounds__(TILE_W) void conv_quad_interp3d_kernel(
    const float* __restrict__ x, float* __restrict__ out) {
  const int t  = threadIdx.x;
  const int w0 = blockIdx.x * TILE_W;
  const int h  = blockIdx.y;
  const int z  = blockIdx.z;          // (b*C + c)*D + d
  const int d  = z & (D_ - 1);
  const int bc = z >> 2;

  // 3 depth slices x 3 rows x (TILE_W + 2) halo columns, edge-replicated
  __shared__ float s[3][3][TILE_W + 2];

  // Async tile fill: global -> LDS directly through the CDNA5 async engine.
#pragma unroll
  for (int dd = -1; dd <= 1; ++dd) {
    const int dcl = min(max(d + dd, 0), D_ - 1);
    const float* pd = x + ((size_t)bc * D_ + dcl) * ((size_t)H_ * W_);
#pragma unroll
    for (int dh = -1; dh <= 1; ++dh) {
      const int hcl = min(max(h + dh, 0), H_ - 1);
      const float* row = pd + (size_t)hcl * W_;
      const int wa = min(max(w0 - 1 + t, 0), W_ - 1);
      async_load_f32_to_lds(row + wa, &s[dd + 1][dh + 1][t]);
      if (t < 2) {
        const int i2 = TILE_W + t;
        const int wb = min(max(w0 - 1 + i2, 0), W_ - 1);
        async_load_f32_to_lds(row + wb, &s[dd + 1][dh + 1][i2]);
      }
    }
  }

  // gfx1250 speculative prefetch of the next H row of the center slice.
  {
    const float* pref = x + ((size_t)bc * D_ + d) * ((size_t)H_ * W_) +
                        (size_t)min(h + 2, H_ - 1) * W_ + w0 + t;
    __builtin_prefetch(pref, 0, 0);
  }

  // Wait for this wave's async LDS writes, then sync the workgroup.
  asm volatile("s_wait_asynccnt 0x0" ::: "memory");
  __syncthreads();

  const int li = t + 1;  // LDS center column for this thread
#define V(dd, dh, dw) s[(dd) + 1][(dh) + 1][li + (dw)]

  const float c = V(0, 0, 0);

  // 3x3x3 NMS (clamped halo == -inf padded max-pool)
  float mx = -__builtin_inff();
#pragma unroll
  for (int i = 0; i < 3; ++i) {
#pragma unroll
    for (int j = 0; j < 3; ++j) {
      mx = fmaxf(mx, s[i][j][li - 1]);
      mx = fmaxf(mx, s[i][j][li]);
      mx = fmaxf(mx, s[i][j][li + 1]);
    }
  }
  const bool nms = (c == mx);

  // First-order gradients (depth kernel flipped)
  const float b0 = 0.5f * (V(0, 0, 1) - V(0, 0, -1));
  const float b1 = 0.5f * (V(0, 1, 0) - V(0, -1, 0));
  const float b2 = 0.5f * (V(-1, 0, 0) - V(1, 0, 0));

  // Hessian entries
  const float dxx = V(0, 0, 1) - 2.0f * c + V(0, 0, -1);
  const float dyy = V(0, 1, 0) - 2.0f * c + V(0, -1, 0);
  const float dss = V(1, 0, 0) - 2.0f * c + V(-1, 0, 0);
  const float dxy = 0.25f * (V(0, -1, -1) - V(0, -1, 1) - V(0, 1, -1) + V(0, 1, 1));
  const float dys = 0.25f * (-V(-1, -1, 0) + V(-1, 1, 0) + V(1, -1, 0) - V(1, 1, 0));
  const float dxs = 0.25f * (-V(-1, 0, -1) + V(-1, 0, 1) + V(1, 0, -1) - V(1, 0, 1));
#undef V

  // Adjugate / det closed-form symmetric 3x3 solve
  const float c00 = dyy * dss - dys * dys;
  const float c01 = dxs * dys - dxy * dss;
  const float c02 = dxy * dys - dxs * dyy;
  const float c11 = dxx * dss - dxs * dxs;
  const float c12 = dxy * dxs - dxx * dys;
  const float c22 = dxx * dyy - dxy * dxy;
  const float det = dxx * c00 + dxy * c01 + dxs * c02;

  const bool  ok     = nms && (det != 0.0f);
  const float invdet = 1.0f / ((det == 0.0f) ? 1.0f : det);

  float d0 = ok ? -((c00 * b0 + c01 * b1 + c02 * b2) * invdet) : 0.0f;
  float d1 = ok ? -((c01 * b0 + c11 * b1 + c12 * b2) * invdet) : 0.0f;
  float d2 = ok ? -((c02 * b0 + c12 * b1 + c22 * b2) * invdet) : 0.0f;

  const bool small =
      fmaxf(fmaxf(fabsf(d0), fabsf(d1)), fabsf(d2)) <= 0.7f;
  d0 = small ? d0 : 0.0f;
  d1 = small ? d1 : 0.0f;
  d2 = small ? d2 : 0.0f;

  const float dE = 0.5f * (b0 * d0 + b1 * d1 + b2 * d2);
  const float y  = c + dE + (ok ? BONUS : 0.0f);

  // Outputs: coords_max (B,C,3,D,H,W) then y_max (B,C,D,H,W).
  // Streamed once, never re-read -> non-temporal stores (TH_NT) to keep the
  // re-read input halo resident in the 192 MB L2.
  const int    w     = w0 + t;
  const size_t hw    = (size_t)h * W_ + w;
  const size_t plane = (size_t)D_ * H_ * W_;                       // k-stride
  const size_t cbase = ((size_t)bc * 3 * D_ + d) * ((size_t)H_ * W_) + hw;

  __builtin_nontemporal_store((float)d + d2,       out + cbase);
  __builtin_nontemporal_store((float)w + d0 + ADJ, out + cbase + plane);
  __builtin_nontemporal_store((float)h + d1 + ADJ, out + cbase + 2 * plane);

  const size_t Ntot = (size_t)B_ * C_ * plane;     // 6,291,456
  __builtin_nontemporal_store(
      y, out + 3 * Ntot + (size_t)z * ((size_t)H_ * W_) + hw);
}

}  // namespace

extern "C" void kernel_launch(void* const* d_in, const int* in_sizes, int n_in,
                              void* d_out, int out_size, void* d_ws, size_t ws_size,
                              hipStream_t stream) {
  (void)in_sizes; (void)n_in; (void)d_ws; (void)ws_size; (void)out_size;
  const float* x  = (const float*)d_in[0];
  float* out      = (float*)d_out;
  dim3 grid(W_ / TILE_W, H_, B_ * C_ * D_);
  dim3 block(TILE_W);
  conv_quad_interp3d_kernel<<<grid, block, 0, stream>>>(x, out);
}